// RDA_44074954392189
// MI455X (gfx1250) — compile-verified
//
#include <hip/hip_runtime.h>
#include <math.h>

#define NA 4096
#define NR 2048
#define TPB 256
#define WAVES (TPB / 32)

static constexpr double PI_D = 3.14159265358979323846;

typedef float v2f __attribute__((ext_vector_type(2)));
typedef float v8f __attribute__((ext_vector_type(8)));
typedef int   v2i __attribute__((ext_vector_type(2)));

// ---- async global->LDS helpers (CDNA5 GLOBAL_LOAD_ASYNC_TO_LDS_*) ---------
#if __has_builtin(__builtin_amdgcn_global_load_async_to_lds_b32) && \
    __has_builtin(__builtin_amdgcn_global_load_async_to_lds_b64)
#define HAVE_ASYNC_LDS 1
#else
#define HAVE_ASYNC_LDS 0
#endif

typedef __attribute__((address_space(1))) int  g_i32;
typedef __attribute__((address_space(3))) int  l_i32;
typedef __attribute__((address_space(1))) v2i  g_v2i;
typedef __attribute__((address_space(3))) v2i  l_v2i;

__device__ __forceinline__ void wait_async0() {
#if __has_builtin(__builtin_amdgcn_s_wait_asynccnt)
    __builtin_amdgcn_s_wait_asynccnt(0);
#else
    asm volatile("s_wait_asynccnt 0" ::: "memory");
#endif
}

// Copy one float2 global -> LDS (async, ASYNCcnt-tracked).
__device__ __forceinline__ void ld_async_f2(float2* lds_dst, const float2* gsrc) {
#if HAVE_ASYNC_LDS
    __builtin_amdgcn_global_load_async_to_lds_b64((g_v2i*)gsrc, (l_v2i*)lds_dst, 0, 0);
#else
    *lds_dst = *gsrc;
#endif
}
// Copy one float global -> LDS (async).
__device__ __forceinline__ void ld_async_f1(float* lds_dst, const float* gsrc) {
#if HAVE_ASYNC_LDS
    __builtin_amdgcn_global_load_async_to_lds_b32((g_i32*)gsrc, (l_i32*)lds_dst, 0, 0);
#else
    *lds_dst = *gsrc;
#endif
}

// ---------------------------------------------------------------------------
// Build LDS twiddle table: tw[j] = (cos, sin)(2*pi*j/N).  One sincosf per
// element per block, amortized over every stage of (both) FFTs.
// ---------------------------------------------------------------------------
template <int N>
__device__ void build_twiddles(float2* tw, int tid) {
    const float step = (float)(2.0 * PI_D) / (float)N;
    for (int j = tid; j < N; j += TPB) {
        float s, c;
        sincosf(step * (float)j, &s, &c);
        tw[j] = make_float2(c, s);
    }
}

// ---------------------------------------------------------------------------
// Stockham FFT of length N in LDS (ping-pong), DIR = -1 fwd, +1 inv.
// Stage 1: radix-16 as a complex 16x16 DFT matmul on WMMA
// (V_WMMA_F32_16X16X4_F32, 4 chained K-chunks per real product, one batch of
// 16 sub-transforms per wave).  Remaining log2(N/16) stages radix-2.
// MODIN folds the (-1)^n centering modulation into the first-stage reads
// (valid because parity(p + (N/16)*r) == parity(p) for even N/16), which lets
// callers stage raw async copies into LDS.  All twiddles from tw[].
// ---------------------------------------------------------------------------
template <int N, int DIR, bool MODIN>
__device__ float2* block_fft(float2* bin, float2* bout, const float2* tw) {
    const int tid  = threadIdx.x;
    const int lane = tid & 31;
    const int wave = tid >> 5;
    constexpr int M16    = N / 16;
    constexpr int NBATCH = M16 / 16;
    constexpr int BPW    = NBATCH / WAVES;

    // ---- radix-16 stage via WMMA -----------------------------------------
    // A = F16 DFT matrix: lanes 0-15 -> (M=lane, K, K+1); 16-31 -> (M, K+2, K+3).
    const int mrow = lane & 15;
    const int kb   = (lane < 16) ? 0 : 2;
    v2f aFr[4], aFi[4];
#pragma unroll
    for (int j = 0; j < 4; ++j) {
        int k0 = 4 * j + kb, k1 = k0 + 1;
        float2 t0 = tw[((mrow * k0) & 15) * M16];   // e^{2pi i (mrow*k0)/16}
        float2 t1 = tw[((mrow * k1) & 15) * M16];
        aFr[j][0] = t0.x;              aFr[j][1] = t1.x;
        aFi[j][0] = (float)DIR * t0.y; aFi[j][1] = (float)DIR * t1.y;
    }
#pragma unroll 1
    for (int b = 0; b < BPW; ++b) {
        const int p = (wave * BPW + b) * 16 + mrow;   // sub-transform (column)
        const float sgp = (MODIN && (p & 1)) ? -1.f : 1.f;
        // B = X data (4x16): lanes 0-15 -> rows (K,K+1); 16-31 -> (K+2,K+3).
        v2f bXr[4], bXi[4], bXiN[4];
#pragma unroll
        for (int j = 0; j < 4; ++j) {
            int r0 = 4 * j + kb, r1 = r0 + 1;
            float2 e0 = bin[p + M16 * r0];
            float2 e1 = bin[p + M16 * r1];
            bXr[j][0]  =  sgp * e0.x; bXr[j][1]  =  sgp * e1.x;
            bXi[j][0]  =  sgp * e0.y; bXi[j][1]  =  sgp * e1.y;
            bXiN[j][0] = -bXi[j][0];  bXiN[j][1] = -bXi[j][1];
        }
        v8f dr = {0.f, 0.f, 0.f, 0.f, 0.f, 0.f, 0.f, 0.f};
        v8f di = {0.f, 0.f, 0.f, 0.f, 0.f, 0.f, 0.f, 0.f};
#pragma unroll
        for (int j = 0; j < 4; ++j)  // Yr = Fr*Xr - Fi*Xi
            dr = __builtin_amdgcn_wmma_f32_16x16x4_f32(false, aFr[j], false, bXr[j], (short)0, dr, false, false);
#pragma unroll
        for (int j = 0; j < 4; ++j)
            dr = __builtin_amdgcn_wmma_f32_16x16x4_f32(false, aFi[j], false, bXiN[j], (short)0, dr, false, false);
#pragma unroll
        for (int j = 0; j < 4; ++j)  // Yi = Fi*Xr + Fr*Xi
            di = __builtin_amdgcn_wmma_f32_16x16x4_f32(false, aFi[j], false, bXr[j], (short)0, di, false, false);
#pragma unroll
        for (int j = 0; j < 4; ++j)
            di = __builtin_amdgcn_wmma_f32_16x16x4_f32(false, aFr[j], false, bXi[j], (short)0, di, false, false);
        // D layout: VGPR i -> t = i (lanes 0-15) / i+8 (lanes 16-31).
        const int tb = (lane < 16) ? 0 : 8;
#pragma unroll
        for (int i = 0; i < 8; ++i) {
            int t = i + tb;
            float2 tt = tw[p * t];                   // W_N^{p t}  (p*t < N)
            float ct = tt.x, st = (float)DIR * tt.y;
            bout[16 * p + t] = make_float2(dr[i] * ct - di[i] * st,
                                           dr[i] * st + di[i] * ct);
        }
    }
    __syncthreads();

    // ---- remaining radix-2 Stockham stages (n*s == N invariant) ----------
    float2* cur = bout;
    float2* nxt = bin;
    int n = M16, s = 16;
    while (n > 1) {
        const int mh = n >> 1;
        for (int w = tid; w < N / 2; w += TPB) {
            int pp = w / s;
            int q  = w - pp * s;
            float2 a  = cur[q + s * pp];
            float2 bb = cur[q + s * (pp + mh)];
            float2 tt = tw[pp * s];                  // W_n^pp = W_N^{pp*s}
            float ct = tt.x, st = (float)DIR * tt.y;
            float dxr = a.x - bb.x, dxi = a.y - bb.y;
            nxt[q + s * (2 * pp)]     = make_float2(a.x + bb.x, a.y + bb.y);
            nxt[q + s * (2 * pp + 1)] = make_float2(dxr * ct - dxi * st,
                                                    dxr * st + dxi * ct);
        }
        __syncthreads();
        float2* tsw = cur; cur = nxt; nxt = tsw;
        n = mh; s <<= 1;
    }
    return cur;
}

// ---------------------------------------------------------------------------
// Kernel A: per-row centered range FFT, * H_rc -> out0 (S_rf_comp),
//           then centered range IFFT -> ws (S_rc).
// ---------------------------------------------------------------------------
__global__ void __launch_bounds__(TPB) range_kernel(const float* __restrict__ sr,
                                                    const float* __restrict__ si,
                                                    float2* __restrict__ out0,
                                                    float2* __restrict__ ws) {
    extern __shared__ float2 smem[];
    float2* b0 = smem;
    float2* b1 = smem + NR;
    float2* tw = smem + 2 * NR;
    const int row = blockIdx.x;
    const int tid = threadIdx.x;
    const float* rr = sr + (size_t)row * NR;
    const float* ri = si + (size_t)row * NR;
    // Stage raw planar input into interleaved LDS via async copies;
    // the (-1)^n modulation is folded into the FFT (MODIN).
    for (int k = tid; k < NR; k += TPB) {
        ld_async_f1(&b0[k].x, rr + k);
        ld_async_f1(&b0[k].y, ri + k);
    }
    build_twiddles<NR>(tw, tid);
    wait_async0();
    __syncthreads();
    float2* res = block_fft<NR, -1, true>(b0, b1, tw);
    float2* oth = (res == b0) ? b1 : b0;

    const float fstep = (float)(32317000.0 / (double)(NR - 1));
    const float fhalf = (float)(32317000.0 * 0.5);
    const float kr    = 721350000000.0f;
    float2* o0 = out0 + (size_t)row * NR;
    for (int k = tid; k < NR; k += TPB) {
        float ft = -fhalf + fstep * (float)k;
        float ph = (float)PI_D * ft * ft / kr;       // H_rc phase
        float s, c; sincosf(ph, &s, &c);
        float2 v = res[k];
        float2 h = make_float2(v.x * c - v.y * s, v.x * s + v.y * c);
        float sg = (k & 1) ? -1.f : 1.f;
        o0[k]  = make_float2(sg * h.x, sg * h.y);    // S_rf_comp = (-1)^k B[k] H_rc
        res[k] = h;                                  // IFFT input: sign pair cancels
    }
    __syncthreads();
    float2* res2 = block_fft<NR, 1, false>(res, oth, tw);
    float2* w = ws + (size_t)row * NR;
    const float invn = 1.0f / (float)NR;
    for (int k = tid; k < NR; k += TPB) {
        float sg = (k & 1) ? -invn : invn;           // (-1)^k / N
        float2 v = res2[k];
        w[k] = make_float2(sg * v.x, sg * v.y);      // S_rc
    }
}

// ---------------------------------------------------------------------------
// Kernels B/D: per-column centered azimuth FFT/IFFT, in place through LDS.
// The 64 MiB array is resident in the 192 MB L2, so the stride-16KB column
// traffic stays on-die; async copies keep the staging out of VGPRs.
// ---------------------------------------------------------------------------
template <int DIR>
__global__ void __launch_bounds__(TPB) az_kernel(float2* ws, float2* out) {
    extern __shared__ float2 smem[];
    float2* b0 = smem;
    float2* b1 = smem + NA;
    float2* tw = smem + 2 * NA;
    const int col = blockIdx.x;
    const int tid = threadIdx.x;
    for (int a = tid; a < NA; a += TPB)
        ld_async_f2(&b0[a], ws + (size_t)a * NR + col);
    build_twiddles<NA>(tw, tid);
    wait_async0();
    __syncthreads();
    float2* res = block_fft<NA, DIR, true>(b0, b1, tw);   // (-1)^a folded in
    const float scale = (DIR < 0) ? 1.0f : (1.0f / (float)NA);
    for (int a = tid; a < NA; a += TPB) {
        float sg = (a & 1) ? -scale : scale;
        float2 v = res[a];
        out[(size_t)a * NR + col] = make_float2(sg * v.x, sg * v.y);
    }
}

// ---------------------------------------------------------------------------
// Kernel C: per-row 6-tap sinc RCMC -> out1 (S_rd_rcmc), then * H_ac back
// into ws (S_rd_ac).  sinc taps use sin(pi(u-off)) = (-1)^off sin(pi u):
// one sinf per sample instead of six.
// ---------------------------------------------------------------------------
__global__ void __launch_bounds__(TPB) rcmc_kernel(float2* ws, float2* __restrict__ out1) {
    extern __shared__ float2 srow[];
    const int row = blockIdx.x;
    const int tid = threadIdx.x;
    float2* w = ws + (size_t)row * NR;
    for (int r = tid; r < NR; r += TPB)
        ld_async_f2(&srow[r], w + r);
    wait_async0();
    __syncthreads();

    const double VRd = 7062.0, LAMd = 0.05667, THC = 0.06786, FAd = 1733.0,
                 FRd = 32317000.0, R0d = 988647.0, CLd = 299790000.0;
    const float feta = (float)(-FAd / 2.0 + (double)row * (FAd / (double)(NA - 1))
                               + 2.0 * VRd * sin(THC) / LAMd);
    const float c1  = (float)(LAMd * LAMd * R0d / (4.0 * CLd * VRd * VRd) * FRd);
    const float dn  = c1 * feta * feta;                       // delta_n(row)
    const float lam = (float)LAMd;
    const float dfe = sqrtf(1.0f - (lam * feta) * (lam * feta) / (float)(4.0 * VRd * VRd));
    const float ph  = (float)(4.0 * PI_D * R0d) * dfe / lam;  // H_ac phase
    float hs, hc; sincosf(ph, &hs, &hc);

    float2* o1 = out1 + (size_t)row * NR;
    for (int r = tid; r < NR; r += TPB) {
        float pos = (float)r + dn;
        float fn0 = floorf(pos);
        float u   = pos - fn0;                                // before clip (as ref)
        int n0 = (int)fn0;
        n0 = min(max(n0, 0), NR - 1);
        float spu = sinf((float)PI_D * u);                    // sin(pi*u), shared
        float ar = 0.f, ai = 0.f;
#pragma unroll
        for (int t = 0; t < 6; ++t) {
            int off = t - 3;                                  // offsets = arange(-3,3)
            int idx = min(max(n0 + off, 0), NR - 1);
            float x  = (float)PI_D * (u - (float)off);
            float sn = (off & 1) ? -spu : spu;                // (-1)^off sin(pi u)
            float wt = (fabsf(x) < 1e-8f) ? 1.0f : (sn / x);  // sinc(u-off)
            float2 sm = srow[idx];
            ar += wt * sm.x; ai += wt * sm.y;
        }
        o1[r] = make_float2(ar, ai);                          // S_rd_rcmc
        w[r]  = make_float2(ar * hc - ai * hs, ar * hs + ai * hc);  // S_rd_ac
    }
}

// ---------------------------------------------------------------------------
extern "C" void kernel_launch(void* const* d_in, const int* in_sizes, int n_in,
                              void* d_out, int out_size, void* d_ws, size_t ws_size,
                              hipStream_t stream) {
    const float* sr = (const float*)d_in[0];
    const float* si = (const float*)d_in[1];
    float2* out  = (float2*)d_out;
    float2* out0 = out;                                 // S_rf_comp
    float2* out1 = out + (size_t)NA * NR;               // S_rd_rcmc
    float2* out2 = out + 2 * (size_t)NA * NR;           // S_image
    float2* ws   = (float2*)d_ws;                       // one 64 MiB complex array

    const size_t lds_r = 3 * (size_t)NR * sizeof(float2);   // 48 KB: ping-pong + tw
    const size_t lds_a = 3 * (size_t)NA * sizeof(float2);   // 96 KB: ping-pong + tw
    const size_t lds_c = (size_t)NR * sizeof(float2);       // 16 KB row

    range_kernel<<<NA, TPB, lds_r, stream>>>(sr, si, out0, ws);  // -> S_rc in ws
    az_kernel<-1><<<NR, TPB, lds_a, stream>>>(ws, ws);           // -> S_rd in ws
    rcmc_kernel<<<NA, TPB, lds_c, stream>>>(ws, out1);           // -> S_rd_ac in ws
    az_kernel<1><<<NR, TPB, lds_a, stream>>>(ws, out2);          // -> S_image
}